// CandidateFilterModel_69286412419195
// MI455X (gfx1250) — compile-verified
//
#include <hip/hip_runtime.h>
#include <math.h>

// Problem constants (match reference)
#define S_LEN  2048
#define HID    1024
#define NHEADS 16
#define NE     256
#define NMENT  1024
#define NP     2048
#define PHID   1024

typedef __attribute__((ext_vector_type(2))) float v2f;
typedef __attribute__((ext_vector_type(4))) float v4f;
typedef __attribute__((ext_vector_type(8))) float v8f;
typedef __attribute__((ext_vector_type(4))) int   v4i;

// CDNA5 async global->LDS path (ASYNCcnt), if this toolchain exposes it.
#if __has_builtin(__builtin_amdgcn_global_load_async_to_lds_b128) && \
    __has_builtin(__builtin_amdgcn_s_wait_asynccnt)
#define CF_ASYNC_LDS 1
#endif

#ifdef CF_ASYNC_LDS
// b128 payload type is int4; src is addrspace(1) ("__device__"), dst addrspace(3).
typedef __attribute__((address_space(1))) v4i cf_g_v4i;
typedef __attribute__((address_space(3))) v4i cf_l_v4i;
__device__ __forceinline__ void cf_async_b128(const float* g, float* l) {
  __builtin_amdgcn_global_load_async_to_lds_b128(
      (cf_g_v4i*)g, (cf_l_v4i*)l, /*offset=*/0, /*cpol=*/0);
}
#endif

// ---------------------------------------------------------------------------
// K0: segment boundaries from sorted entity_ids (lower_bound per entity)
// ---------------------------------------------------------------------------
__global__ __launch_bounds__(256)
void seg_bounds_kernel(const int* __restrict__ entity_ids, int* __restrict__ seg_start) {
  int e = blockIdx.x * blockDim.x + threadIdx.x;
  if (e > NE) return;
  int lo = 0, hi = NMENT;
  while (lo < hi) {
    int mid = (lo + hi) >> 1;
    if (entity_ids[mid] < e) lo = mid + 1; else hi = mid;
  }
  seg_start[e] = lo;
}

// ---------------------------------------------------------------------------
// K1: entity embeddings = segment logsumexp over gathered token rows
// ---------------------------------------------------------------------------
__global__ __launch_bounds__(256)
void ent_emb_kernel(const float* __restrict__ seq, const int* __restrict__ mention_idx,
                    const int* __restrict__ seg_start, float* __restrict__ ent_emb) {
  int e  = blockIdx.x;
  int m0 = seg_start[e], m1 = seg_start[e + 1];
  for (int d = threadIdx.x; d < HID; d += blockDim.x) {
    if (m1 <= m0) { ent_emb[(size_t)e * HID + d] = 0.0f; continue; }
    float mx = -3.402823466e38f;
    for (int m = m0; m < m1; ++m) {
      float v = seq[(size_t)mention_idx[m] * HID + d];
      mx = fmaxf(mx, v);
    }
    float s = 0.0f;
    for (int m = m0; m < m1; ++m) {
      float v = seq[(size_t)mention_idx[m] * HID + d];
      s += expf(v - mx);
    }
    ent_emb[(size_t)e * HID + d] = mx + logf(s);
  }
}

// ---------------------------------------------------------------------------
// K2: entity attention = segment mean of gathered attention rows
// grid = NE * NHEADS
// ---------------------------------------------------------------------------
__global__ __launch_bounds__(256)
void ent_att_kernel(const float* __restrict__ att, const int* __restrict__ mention_idx,
                    const int* __restrict__ seg_start, float* __restrict__ ent_att) {
  int e = blockIdx.x / NHEADS;
  int h = blockIdx.x % NHEADS;
  int m0 = seg_start[e], m1 = seg_start[e + 1];
  float inv = 1.0f / fmaxf((float)(m1 - m0), 1.0f);
  const float* base = att + (size_t)h * S_LEN * S_LEN;
  float* out = ent_att + ((size_t)e * NHEADS + h) * S_LEN;
  for (int s = threadIdx.x; s < S_LEN; s += blockDim.x) {
    float acc = 0.0f;
    for (int m = m0; m < m1; ++m)
      acc += base[(size_t)mention_idx[m] * S_LEN + s];
    out[s] = acc * inv;
  }
}

// ---------------------------------------------------------------------------
// K3: pair attention: head-contracted product of two entity attentions,
//     row-normalized.  One block per pair; row kept in registers.
// ---------------------------------------------------------------------------
__global__ __launch_bounds__(256)
void pair_att_kernel(const int* __restrict__ pair_h, const int* __restrict__ pair_t,
                     const float* __restrict__ ent_att, float* __restrict__ pair_att) {
  constexpr int PERT = S_LEN / 256;  // 8 columns per thread
  int p = blockIdx.x;
  int tid = threadIdx.x;
  const float* A = ent_att + (size_t)pair_h[p] * NHEADS * S_LEN;
  const float* B = ent_att + (size_t)pair_t[p] * NHEADS * S_LEN;
  float acc[PERT];
  float tot = 0.0f;
#pragma unroll
  for (int i = 0; i < PERT; ++i) {
    int s = i * 256 + tid;
    float a = 0.0f;
#pragma unroll
    for (int h = 0; h < NHEADS; ++h)
      a += A[h * S_LEN + s] * B[h * S_LEN + s];
    acc[i] = a;
    tot += a;
  }
  __shared__ float red[256];
  red[tid] = tot;
  __syncthreads();
  for (int w = 128; w > 0; w >>= 1) {
    if (tid < w) red[tid] += red[tid + w];
    __syncthreads();
  }
  float inv = 1.0f / (red[0] + 1e-6f);
#pragma unroll
  for (int i = 0; i < PERT; ++i)
    pair_att[(size_t)p * S_LEN + i * 256 + tid] = acc[i] * inv;
}

// ---------------------------------------------------------------------------
// Inner WMMA tile FMA on one staged (BK=16) LDS tile.
//   V_WMMA_F32_16X16X4_F32 fragment layouts (ISA 7.12.2):
//     A: lane L holds (M=L%16, K=kk*4 + 2*(L/16)+v), v=0..1  -> float2
//     B: lane L holds (K=kk*4 + 2*(L/16)+v, N=L%16)
//     C: VGPR r, lane L -> (M = r + 8*(L/16), N = L%16)
// ---------------------------------------------------------------------------
__device__ __forceinline__ void wmma_tile_fma(const float (*__restrict__ Asb)[20],
                                              const float (*__restrict__ Bsb)[68],
                                              int waveM, int waveN, int half, int lm,
                                              v8f (&c)[2][2]) {
#pragma unroll
  for (int kk = 0; kk < 4; ++kk) {
    const int kc = kk * 4 + 2 * half;
    v2f a[2], b[2];
#pragma unroll
    for (int mt = 0; mt < 2; ++mt) {
      const float* ap = &Asb[waveM + mt * 16 + lm][kc];
      a[mt] = v2f{ap[0], ap[1]};
    }
#pragma unroll
    for (int nt = 0; nt < 2; ++nt) {
      const int nn = waveN + nt * 16 + lm;
      b[nt] = v2f{Bsb[kc][nn], Bsb[kc + 1][nn]};
    }
#pragma unroll
    for (int mt = 0; mt < 2; ++mt)
#pragma unroll
      for (int nt = 0; nt < 2; ++nt)
        c[mt][nt] = __builtin_amdgcn_wmma_f32_16x16x4_f32(
            false, a[mt], false, b[nt], (short)0, c[mt][nt], false, false);
  }
}

// ---------------------------------------------------------------------------
// K4: tiled f32 WMMA GEMM:  C[M,N] = act(gather(A)[M,K] @ B[K,N] + bias + add)
//   Block tile 128x64, 8 waves in 4x2, each wave 2x2 tiles of 16x16.
//   BK=16 staged in LDS (A stride 20 floats -> conflict-free column reads).
//   Async path: GLOBAL_LOAD_ASYNC_TO_LDS_B128, double-buffered; last tile
//   peeled so the steady-state loop is branch-free:
//     issue(t+1) -> s_wait_asynccnt 3 -> barrier -> 16x WMMA -> barrier
// ---------------------------------------------------------------------------
template <bool HAS_GATHER, bool HAS_BIAS, bool HAS_ADD, bool DO_TANH>
__global__ __launch_bounds__(256)
void wmma_gemm_f32(const float* __restrict__ A, int lda,
                   const float* __restrict__ B, int ldb,
                   float* __restrict__ C, int ldc, int K,
                   const int* __restrict__ gidx,
                   const float* __restrict__ bias,
                   const float* __restrict__ addmat) {
#ifdef CF_ASYNC_LDS
  __shared__ float As[2][128][20];   // stride 20: 20*r mod 64 distinct, r=0..15
  __shared__ float Bs[2][16][68];
#else
  __shared__ float As[1][128][20];
  __shared__ float Bs[1][16][68];
#endif

  const int tid   = threadIdx.x;
  const int lane  = tid & 31;
  const int wave  = tid >> 5;
  const int waveM = (wave & 3) * 32;
  const int waveN = (wave >> 2) * 32;
  const int half  = lane >> 4;
  const int lm    = lane & 15;
  const int mBase = blockIdx.y * 128;
  const int nBase = blockIdx.x * 64;

  // --- A tile loader: 512 float4 slots, 2 per thread; gather row sources once
  int aR[2], aC4[2];
  const float* aRow[2];
#pragma unroll
  for (int j = 0; j < 2; ++j) {
    int slot = j * 256 + tid;
    aR[j]  = slot >> 2;
    aC4[j] = (slot & 3) * 4;
    int gm  = mBase + aR[j];
    int src = HAS_GATHER ? gidx[gm] : gm;
    aRow[j] = A + (size_t)src * lda + aC4[j];
  }
  // --- B tile loader: 256 float4 slots, 1 per thread
  const int bR  = tid >> 4;
  const int bC4 = (tid & 15) * 4;
  const float* bPtr = B + (size_t)bR * ldb + nBase + bC4;

  v8f c[2][2];
#pragma unroll
  for (int mt = 0; mt < 2; ++mt)
#pragma unroll
    for (int nt = 0; nt < 2; ++nt)
      c[mt][nt] = (v8f)0.0f;

#ifdef CF_ASYNC_LDS
  // -------- async double-buffered pipeline (3 async ops / wave / tile) -----
  const int ntiles = K >> 4;
  cf_async_b128(aRow[0], &As[0][aR[0]][aC4[0]]);
  cf_async_b128(aRow[1], &As[0][aR[1]][aC4[1]]);
  cf_async_b128(bPtr,    &Bs[0][bR][bC4]);
  // steady state: branch-free issue/wait/compute
  for (int t = 0; t < ntiles - 1; ++t) {
    const int buf = t & 1;
    const int k1  = (t + 1) << 4;
    cf_async_b128(aRow[0] + k1, &As[buf ^ 1][aR[0]][aC4[0]]);
    cf_async_b128(aRow[1] + k1, &As[buf ^ 1][aR[1]][aC4[1]]);
    cf_async_b128(bPtr + (size_t)k1 * ldb, &Bs[buf ^ 1][bR][bC4]);
    __builtin_amdgcn_s_wait_asynccnt(3);     // tile t's 3 ops done (in-order)
    __syncthreads();                         // all waves' data landed in LDS
    wmma_tile_fma(As[buf], Bs[buf], waveM, waveN, half, lm, c);
    __syncthreads();                         // done reading before buf reuse
  }
  // peeled last tile
  {
    const int buf = (ntiles - 1) & 1;
    __builtin_amdgcn_s_wait_asynccnt(0);
    __syncthreads();
    wmma_tile_fma(As[buf], Bs[buf], waveM, waveN, half, lm, c);
  }
#else
  // -------- synchronous fallback: hoist all loads before LDS stores --------
  for (int k0 = 0; k0 < K; k0 += 16) {
    v4f va0 = *(const v4f*)(aRow[0] + k0);
    v4f va1 = *(const v4f*)(aRow[1] + k0);
    v4f vb  = *(const v4f*)(bPtr + (size_t)k0 * ldb);
    if (k0 + 16 < K) {
      __builtin_prefetch(aRow[0] + k0 + 16, 0, 0);
      __builtin_prefetch(aRow[1] + k0 + 16, 0, 0);
      __builtin_prefetch(bPtr + (size_t)(k0 + 16) * ldb, 0, 0);
    }
    __syncthreads();   // previous tile fully consumed
    As[0][aR[0]][aC4[0] + 0] = va0[0];
    As[0][aR[0]][aC4[0] + 1] = va0[1];
    As[0][aR[0]][aC4[0] + 2] = va0[2];
    As[0][aR[0]][aC4[0] + 3] = va0[3];
    As[0][aR[1]][aC4[1] + 0] = va1[0];
    As[0][aR[1]][aC4[1] + 1] = va1[1];
    As[0][aR[1]][aC4[1] + 2] = va1[2];
    As[0][aR[1]][aC4[1] + 3] = va1[3];
    *(v4f*)&Bs[0][bR][bC4] = vb;   // 272B row stride keeps 16B alignment
    __syncthreads();
    wmma_tile_fma(As[0], Bs[0], waveM, waveN, half, lm, c);
  }
#endif

  // --- epilogue: bias + ctx add + tanh, then store
#pragma unroll
  for (int mt = 0; mt < 2; ++mt)
#pragma unroll
    for (int nt = 0; nt < 2; ++nt) {
      const int gn = nBase + waveN + nt * 16 + lm;
      float bv = HAS_BIAS ? bias[gn] : 0.0f;
#pragma unroll
      for (int r = 0; r < 8; ++r) {
        const int gm = mBase + waveM + mt * 16 + half * 8 + r;
        float v = c[mt][nt][r];
        if (HAS_BIAS) v += bv;
        if (HAS_ADD)  v += addmat[(size_t)gm * ldc + gn];
        if (DO_TANH)  v = tanhf(v);
        C[(size_t)gm * ldc + gn] = v;
      }
    }
}

// ---------------------------------------------------------------------------
// K5: logits[p] = dot(tbil[p,:], z_o[p,:]) + b_bil
// ---------------------------------------------------------------------------
__global__ __launch_bounds__(256)
void logits_kernel(const float* __restrict__ tbil, const float* __restrict__ z_o,
                   const float* __restrict__ b_bil, float* __restrict__ out) {
  int p = blockIdx.x, tid = threadIdx.x;
  float acc = 0.0f;
  for (int j = tid; j < PHID; j += 256)
    acc += tbil[(size_t)p * PHID + j] * z_o[(size_t)p * PHID + j];
  __shared__ float red[256];
  red[tid] = acc;
  __syncthreads();
  for (int w = 128; w > 0; w >>= 1) {
    if (tid < w) red[tid] += red[tid + w];
    __syncthreads();
  }
  if (tid == 0) out[p] = red[0] + b_bil[0];
}

// ---------------------------------------------------------------------------
extern "C" void kernel_launch(void* const* d_in, const int* in_sizes, int n_in,
                              void* d_out, int out_size, void* d_ws, size_t ws_size,
                              hipStream_t stream) {
  const float* seq         = (const float*)d_in[0];
  const float* att         = (const float*)d_in[1];
  const int*   mention_idx = (const int*)d_in[2];
  const int*   entity_ids  = (const int*)d_in[3];
  const int*   pair_h      = (const int*)d_in[4];
  const int*   pair_t      = (const int*)d_in[5];
  const float* W_head      = (const float*)d_in[6];
  const float* b_head      = (const float*)d_in[7];
  const float* W_tail      = (const float*)d_in[8];
  const float* b_tail      = (const float*)d_in[9];
  const float* W_ctx       = (const float*)d_in[10];
  const float* W_bil       = (const float*)d_in[11];
  const float* b_bil       = (const float*)d_in[12];
  float* out = (float*)d_out;

  // workspace carve-up (~81 MB)
  char* ws = (char*)d_ws;
  int* seg_start = (int*)ws;                  // (NE+1) ints, padded to 4KB
  float* fws = (float*)(ws + 4096);
  float* ent_emb  = fws;  fws += (size_t)NE * HID;             // 1 MB
  float* ent_att  = fws;  fws += (size_t)NE * NHEADS * S_LEN;  // 32 MB
  float* pair_att = fws;  fws += (size_t)NP * S_LEN;           // 16 MB
  float* contexts = fws;  fws += (size_t)NP * HID;             // 8 MB
  float* ctx_p    = fws;  fws += (size_t)NP * PHID;            // 8 MB
  float* z_s      = fws;  fws += (size_t)NP * PHID;            // 8 MB
  float* z_o      = fws;  fws += (size_t)NP * PHID;            // 8 MB
  float* tbil     = pair_att;  // pair_att dead after `contexts`; reuse

  seg_bounds_kernel<<<(NE + 1 + 255) / 256, 256, 0, stream>>>(entity_ids, seg_start);
  ent_emb_kernel<<<NE, 256, 0, stream>>>(seq, mention_idx, seg_start, ent_emb);
  ent_att_kernel<<<NE * NHEADS, 256, 0, stream>>>(att, mention_idx, seg_start, ent_att);
  pair_att_kernel<<<NP, 256, 0, stream>>>(pair_h, pair_t, ent_att, pair_att);

  dim3 blk(256);
  dim3 gCtx(HID / 64, NP / 128);   // contexts: N = HID
  dim3 gPh(PHID / 64, NP / 128);   // projections: N = PHID

  // contexts = pair_att @ seq                    (M=2048, K=2048, N=1024)
  wmma_gemm_f32<false, false, false, false><<<gCtx, blk, 0, stream>>>(
      pair_att, S_LEN, seq, HID, contexts, HID, S_LEN, nullptr, nullptr, nullptr);
  // ctx_p = contexts @ W_ctx                     (K=1024)
  wmma_gemm_f32<false, false, false, false><<<gPh, blk, 0, stream>>>(
      contexts, HID, W_ctx, PHID, ctx_p, PHID, HID, nullptr, nullptr, nullptr);
  // z_s = tanh(ent_emb[pair_h] @ W_head + b_head + ctx_p)
  wmma_gemm_f32<true, true, true, true><<<gPh, blk, 0, stream>>>(
      ent_emb, HID, W_head, PHID, z_s, PHID, HID, pair_h, b_head, ctx_p);
  // z_o = tanh(ent_emb[pair_t] @ W_tail + b_tail + ctx_p)
  wmma_gemm_f32<true, true, true, true><<<gPh, blk, 0, stream>>>(
      ent_emb, HID, W_tail, PHID, z_o, PHID, HID, pair_t, b_tail, ctx_p);
  // tbil = z_s @ W_bil                           (K=1024)
  wmma_gemm_f32<false, false, false, false><<<gPh, blk, 0, stream>>>(
      z_s, PHID, W_bil, PHID, tbil, PHID, PHID, nullptr, nullptr, nullptr);

  logits_kernel<<<NP, 256, 0, stream>>>(tbil, z_o, b_bil, out);
}